// Attention_39702677684670
// MI455X (gfx1250) — compile-verified
//
#include <hip/hip_runtime.h>
#include <hip/hip_bf16.h>

// Problem constants (reference: N=128, T=2048, D=512)
#define NB 128
#define TT 2048
#define DD 512
#define NSLICE 4            // T-split for the A*V phase (parallelism + determinism)
#define NEGV (-1.0e9f)

typedef float v2f __attribute__((ext_vector_type(2)));
typedef float v8f __attribute__((ext_vector_type(8)));

// ---------------------------------------------------------------------------
// Kernel 1: energy = K[n] . q[n] via V_WMMA_F32_16X16X4_F32 (q broadcast over
// the 16 columns), then masked softmax in LDS; writes attention weights to
// workspace and the boolean mask (as 0.0/1.0 float) to d_out.
// Grid: NB blocks x 256 threads (8 waves). Each wave owns 16 t-rows per pass.
// ---------------------------------------------------------------------------
__global__ __launch_bounds__(256) void attn_energy_softmax(
    const float* __restrict__ q, const float* __restrict__ K,
    const int* __restrict__ lens, float* __restrict__ att,
    float* __restrict__ mask_out)
{
    __shared__ float q_lds[DD];     // 2 KB
    __shared__ float e_lds[TT];     // 8 KB
    __shared__ float red[8];

    const int n    = blockIdx.x;
    const int tid  = threadIdx.x;
    const int lane = tid & 31;
    const int wave = tid >> 5;      // 0..7
    const int half = lane >> 4;     // K placement: VGPR0 <-> K=2*half, VGPR1 <-> K=2*half+1
    const int l15  = lane & 15;     // A: row M;  B: col N

    for (int i = tid; i < DD; i += 256) q_lds[i] = q[(size_t)n * DD + i];
    __syncthreads();

    const float* Kn = K + (size_t)n * TT * DD;

    // 16 passes x (8 waves * 16 rows) = 2048 t's
    for (int pass = 0; pass < TT / 128; ++pass) {
        const int t0 = pass * 128 + wave * 16;
        const float* krow = Kn + (size_t)(t0 + l15) * DD + half * 2;
        v8f acc = {};
        #pragma unroll 8
        for (int d0 = 0; d0 < DD; d0 += 4) {
            v2f a = *(const v2f*)(krow + d0);              // A: K[t0+l15, d0+2h .. +1]
            v2f b = *(const v2f*)(&q_lds[d0 + half * 2]);  // B: q broadcast across 16 cols
            acc = __builtin_amdgcn_wmma_f32_16x16x4_f32(
                false, a, false, b, (short)0, acc, false, false);
        }
        // D[m,j] identical for all j; VGPR r holds rows r (lanes 0-15) and r+8 (16-31)
        if (l15 == 0) {
            #pragma unroll
            for (int r = 0; r < 8; ++r) e_lds[t0 + half * 8 + r] = acc[r];
        }
    }
    __syncthreads();

    const int len = lens[n];

    // masked max
    float lmax = -3.4e38f;
    for (int t = tid; t < TT; t += 256)
        lmax = fmaxf(lmax, (t < len) ? e_lds[t] : NEGV);
    for (int off = 16; off > 0; off >>= 1)
        lmax = fmaxf(lmax, __shfl_xor(lmax, off, 32));
    if (lane == 0) red[wave] = lmax;
    __syncthreads();
    float mx = red[0];
    #pragma unroll
    for (int wv = 1; wv < 8; ++wv) mx = fmaxf(mx, red[wv]);

    // exp + sum (each thread owns a strided slice of e_lds -> safe in-place)
    float lsum = 0.0f;
    for (int t = tid; t < TT; t += 256) {
        float w = (t < len) ? __expf(e_lds[t] - mx) : 0.0f;
        e_lds[t] = w;
        lsum += w;
    }
    for (int off = 16; off > 0; off >>= 1)
        lsum += __shfl_xor(lsum, off, 32);
    __syncthreads();                 // red[] reuse
    if (lane == 0) red[wave] = lsum;
    __syncthreads();
    float sum = 0.0f;
    #pragma unroll
    for (int wv = 0; wv < 8; ++wv) sum += red[wv];
    const float inv = 1.0f / sum;

    for (int t = tid; t < TT; t += 256) {
        att[(size_t)n * TT + t]      = e_lds[t] * inv;
        mask_out[(size_t)n * TT + t] = (t >= len) ? 1.0f : 0.0f;
    }
}

// ---------------------------------------------------------------------------
// Kernel 2: partial context = a[slice] . V[n, slice, :] via WMMA (attention
// weights broadcast over the 16 rows). Grid: NB*NSLICE blocks x 256 threads.
// Each wave owns 4 d-chunks of 16; loops over its 512-long t slice.
// ---------------------------------------------------------------------------
__global__ __launch_bounds__(256) void attn_av_partial(
    const float* __restrict__ V, const float* __restrict__ att,
    float* __restrict__ partial)
{
    __shared__ float a_lds[TT / NSLICE];   // 2 KB

    const int n     = blockIdx.x >> 2;
    const int s     = blockIdx.x & (NSLICE - 1);
    const int tid   = threadIdx.x;
    const int lane  = tid & 31;
    const int wave  = tid >> 5;
    const int half  = lane >> 4;
    const int l15   = lane & 15;
    const int tbase = s * (TT / NSLICE);

    for (int i = tid; i < TT / NSLICE; i += 256)
        a_lds[i] = att[(size_t)n * TT + tbase + i];
    __syncthreads();

    const float* Vn = V + (size_t)n * TT * DD;

    #pragma unroll
    for (int c = 0; c < 4; ++c) {
        const int d0 = (wave * 4 + c) * 16;
        v8f acc = {};
        #pragma unroll 4
        for (int t = 0; t < TT / NSLICE; t += 4) {
            v2f a = *(const v2f*)(&a_lds[t + half * 2]);   // A: a broadcast across 16 rows
            const float* vp = Vn + (size_t)(tbase + t + half * 2) * DD + d0 + l15;
            v2f b;
            b[0] = vp[0];                                  // B: V[t+2h  , d0+l15]
            b[1] = vp[DD];                                 // B: V[t+2h+1, d0+l15]
            acc = __builtin_amdgcn_wmma_f32_16x16x4_f32(
                false, a, false, b, (short)0, acc, false, false);
        }
        // every row of D identical; row 0 = lanes 0-15 of acc[0]
        if (lane < 16)
            partial[((size_t)s * NB + n) * DD + d0 + lane] = acc[0];
    }
}

// ---------------------------------------------------------------------------
// Kernel 3: context[n,d] = sum over slices (fixed order -> deterministic)
// ---------------------------------------------------------------------------
__global__ __launch_bounds__(256) void attn_reduce_ctx(
    const float* __restrict__ partial, float* __restrict__ ctx)
{
    const int i = blockIdx.x * 256 + threadIdx.x;
    if (i < NB * DD) {
        float s0 = partial[i];
        float s1 = partial[1 * NB * DD + i];
        float s2 = partial[2 * NB * DD + i];
        float s3 = partial[3 * NB * DD + i];
        ctx[i] = (s0 + s1) + (s2 + s3);
    }
}

extern "C" void kernel_launch(void* const* d_in, const int* in_sizes, int n_in,
                              void* d_out, int out_size, void* d_ws, size_t ws_size,
                              hipStream_t stream) {
    const float* q    = (const float*)d_in[0];   // [N, D]
    const float* K    = (const float*)d_in[1];   // [N, T, D]
    const float* V    = (const float*)d_in[2];   // [N, T, D]
    const int*   lens = (const int*)d_in[3];     // [N]

    float* out   = (float*)d_out;
    float* ctx   = out;                          // [N, D]
    float* maskf = out + (size_t)NB * DD;        // [N, T] as 0.0/1.0

    float* att     = (float*)d_ws;                         // [N, T]      (1 MB)
    float* partial = att + (size_t)NB * TT;                // [NSLICE,N,D] (1 MB)

    attn_energy_softmax<<<NB, 256, 0, stream>>>(q, K, lens, att, maskf);
    attn_av_partial<<<NB * NSLICE, 256, 0, stream>>>(V, att, partial);
    attn_reduce_ctx<<<(NB * DD) / 256, 256, 0, stream>>>(partial, ctx);
}